// HaarWaveletLayer_23648089932287
// MI455X (gfx1250) — compile-verified
//
#include <hip/hip_runtime.h>
#include <hip/hip_bf16.h>
#include <stddef.h>

// Fused Haar-DWT2 + 1x1-conv ("fusion") for MI455X (gfx1250, wave32).
//
// out[b,o,i,j] = sum_{c=0..2} sum_{q=0..3} We[o, c*4+q] * quad_q(b,c,i,j)
// where quad = (a,b,c,d) = 2x2 input block and We folds the +-0.5 Haar
// butterfly into the 64x12 conv weights. Single pass: read x once (100MB),
// write out once (512MB) -> ~26us bandwidth floor @ 23.3 TB/s. The K=12 dot
// products run on the matrix pipe via V_WMMA_F32_16X16X4_F32 (full fp32).

typedef __attribute__((ext_vector_type(2))) float v2f;
typedef __attribute__((ext_vector_type(8))) float v8f;

// Effective (Haar-folded) weight for quad position q in {0:a,1:b,2:c,3:d}.
// band weights w4 = (wLL, wLH, wHL, wHH) for one input channel.
__device__ __forceinline__ float eff_w(const float4 w4, int q) {
    const float s1 = (q < 2)               ? 1.0f : -1.0f;  // LH: +,+,-,-
    const float s2 = ((q & 1) == 0)        ? 1.0f : -1.0f;  // HL: +,-,+,-
    const float s3 = (q == 0 || q == 3)    ? 1.0f : -1.0f;  // HH: +,-,-,+
    return 0.5f * (w4.x + s1 * w4.y + s2 * w4.z + s3 * w4.w);
}

__global__ __launch_bounds__(256)
void haar_fused_wmma_kernel(const float* __restrict__ x,   // [B,3,512,512]
                            const float* __restrict__ w,   // [64,12]
                            float* __restrict__ out)       // [B,64,256,256]
{
    constexpr int C  = 3;
    constexpr int W  = 512;
    constexpr int H2 = 256;
    constexpr int W2 = 256;
    constexpr int O  = 64;

    const int bi = blockIdx.x / H2;   // batch index
    const int i  = blockIdx.x % H2;   // output row
    const int wave    = threadIdx.x >> 5;
    const int lane    = threadIdx.x & 31;
    const int lane_lo = lane & 15;    // N / M position within half-wave
    const int kh      = lane >> 4;    // lane-half selects K in {0,1} vs {2,3}

    // ---- Per-wave A fragments: effective weights, WMMA f32 16x16x4 A layout.
    // A VGPR0 = A[M = lane_lo (+tile), K = kh*2], VGPR1 = K = kh*2+1.
    v2f A[4][C];
#pragma unroll
    for (int ot = 0; ot < 4; ++ot) {
        const int M = ot * 16 + lane_lo;
#pragma unroll
        for (int c = 0; c < C; ++c) {
            const float4 w4 = *(const float4*)(w + M * 12 + c * 4);
            const int q = kh * 2;
            v2f a;
            a.x = eff_w(w4, q);
            a.y = eff_w(w4, q + 1);
            A[ot][c] = a;
        }
    }

    const int jbase = wave * 32;      // 8 waves x 32 cols = 256 cols

    // Prefetch the two x rows this wave will stream (global_prefetch_b8).
#pragma unroll
    for (int c = 0; c < C; ++c) {
        const float* xp = x + (((size_t)(bi * C + c) * W) + (size_t)(2 * i + kh)) * W
                            + 2 * (jbase + lane_lo);
        __builtin_prefetch(xp, 0, 0);
        __builtin_prefetch(xp + 32, 0, 0);
    }

#pragma unroll
    for (int t = 0; t < 2; ++t) {     // two 16-pixel N-tiles per wave
        const int j = jbase + t * 16 + lane_lo;

        // ---- B fragments: 4x16 f32 B layout.
        // lanes 0-15 hold K=0,1 (quad a,b = row 2i), lanes 16-31 hold K=2,3
        // (quad c,d = row 2i+1) -> one coalesced float2 load per channel.
        v2f Bv[C];
#pragma unroll
        for (int c = 0; c < C; ++c) {
            const float* xp = x + (((size_t)(bi * C + c) * W) + (size_t)(2 * i + kh)) * W
                                + (size_t)(2 * j);
            Bv[c] = *(const v2f*)xp;
        }

        // ---- 4 output-channel tiles x 3 K-steps of WMMA, fp32 accumulate.
        v8f acc[4];
#pragma unroll
        for (int ot = 0; ot < 4; ++ot) {
            v8f a = {};  // C = 0
#pragma unroll
            for (int c = 0; c < C; ++c) {
                a = __builtin_amdgcn_wmma_f32_16x16x4_f32(
                        /*neg_a=*/false, A[ot][c],
                        /*neg_b=*/false, Bv[c],
                        /*c_mod=*/(short)0, a,
                        /*reuse_a=*/false, /*reuse_b=*/false);
            }
            acc[ot] = a;
        }

        // ---- Store D: lane L -> N = L%16; VGPR r -> M = r + 8*(L/16).
        // Each half-wave writes 64 contiguous bytes per (m, r).
#pragma unroll
        for (int ot = 0; ot < 4; ++ot) {
#pragma unroll
            for (int r = 0; r < 8; ++r) {
                const int m = ot * 16 + kh * 8 + r;
                out[(((size_t)bi * O + m) * H2 + i) * W2 + j] = acc[ot][r];
            }
        }
    }
}

extern "C" void kernel_launch(void* const* d_in, const int* in_sizes, int n_in,
                              void* d_out, int out_size, void* d_ws, size_t ws_size,
                              hipStream_t stream) {
    const float* x  = (const float*)d_in[0];   // [B,3,512,512] fp32
    const float* wf = (const float*)d_in[1];   // [64,12,1,1]   fp32
    float* out = (float*)d_out;                // [B,64,256,256] fp32

    const int B = in_sizes[0] / (3 * 512 * 512);   // 32
    dim3 grid(B * 256);   // one block per (batch, output row)
    dim3 block(256);      // 8 wave32 waves; each wave covers 32 columns
    hipLaunchKernelGGL(haar_fused_wmma_kernel, grid, block, 0, stream, x, wf, out);
}